// LightGCN_67757404061704
// MI455X (gfx1250) — compile-verified
//
#include <hip/hip_runtime.h>
#include <math.h>

#define NUM_USERS 100000
#define NUM_ITEMS 200000
#define NUM_NODES 300000
#define DIM 64
#define N64 (NUM_NODES * DIM) /* 19,200,000 floats */
#define NLAYERS 3
#define BATCH 8192

typedef __attribute__((ext_vector_type(2))) float v2f;
typedef __attribute__((ext_vector_type(8))) float v8f;

// acc = cur = concat(user_emb, item_emb), float4-vectorized
__global__ void lgcn_init(const float* __restrict__ ue, const float* __restrict__ ie,
                          float* __restrict__ acc, float* __restrict__ cur) {
    unsigned i = blockIdx.x * blockDim.x + threadIdx.x;   // float4 index
    if (i >= (unsigned)(N64 / 4)) return;
    const unsigned uq = (unsigned)NUM_USERS * DIM / 4;
    float4 v = (i < uq) ? ((const float4*)ue)[i] : ((const float4*)ie)[i - uq];
    ((float4*)acc)[i] = v;
    ((float4*)cur)[i] = v;
}

// 16 lanes per edge, float4 per lane: nxt[dst] += val * cur[src]
__global__ void lgcn_scatter(const int* __restrict__ src, const int* __restrict__ dst,
                             const float* __restrict__ vals,
                             const float* __restrict__ cur, float* __restrict__ nxt,
                             int E) {
    unsigned tid = blockIdx.x * blockDim.x + threadIdx.x;
    unsigned e = tid >> 4;
    if (e >= (unsigned)E) return;
    unsigned c = (tid & 15u) << 2;
    int s = src[e];
    int d = dst[e];
    float v = vals[e];
    float4 x = *(const float4*)(cur + (size_t)s * DIM + c);
    float* o = nxt + (size_t)d * DIM + c;
    atomicAdd(o + 0, v * x.x);
    atomicAdd(o + 1, v * x.y);
    atomicAdd(o + 2, v * x.z);
    atomicAdd(o + 3, v * x.w);
}

// acc += nxt (float4-vectorized)
__global__ void lgcn_accum(float* __restrict__ acc, const float* __restrict__ nxt) {
    unsigned i = blockIdx.x * blockDim.x + threadIdx.x;
    if (i >= (unsigned)(N64 / 4)) return;
    float4 a = ((float4*)acc)[i];
    float4 b = ((const float4*)nxt)[i];
    a.x += b.x; a.y += b.y; a.z += b.z; a.w += b.w;
    ((float4*)acc)[i] = a;
}

// One wave per 16-row batch tile. S = U(16x64) x P^T(64x16) chained through
// 16 V_WMMA_F32_16X16X4_F32 ops (full fp32). A(16x4,M-striped) and B(4x16,
// N-striped) have identical per-lane fetch patterns: lane L reads
// row[L&15][k0 + 2*(L>>4)] as float2. Diagonal of D lives one element per
// lane in lanes 0-7 (vgpr r=N) and 24-31 (r=N-8).
__global__ void lgcn_score(const float* __restrict__ acc,
                           const int* __restrict__ users,
                           const int* __restrict__ pos,
                           const int* __restrict__ neg,
                           float* __restrict__ partials) {
    const int tile = blockIdx.x;
    const int lane = threadIdx.x;
    const int m = lane & 15;
    const int hi = lane >> 4;
    const int bidx = tile * 16 + m;

    const float* uRow = acc + (size_t)users[bidx] * DIM;
    const float* pRow = acc + (size_t)(NUM_USERS + pos[bidx]) * DIM;
    const float* nRow = acc + (size_t)(NUM_USERS + neg[bidx]) * DIM;
    const int koff = 2 * hi;

    v8f cp = {};
    v8f cn = {};
#pragma unroll
    for (int k0 = 0; k0 < DIM; k0 += 4) {
        v2f a  = *(const v2f*)(uRow + k0 + koff);
        v2f bp = *(const v2f*)(pRow + k0 + koff);
        v2f bn = *(const v2f*)(nRow + k0 + koff);
        cp = __builtin_amdgcn_wmma_f32_16x16x4_f32(false, a, false, bp, (short)0, cp, false, false);
        cn = __builtin_amdgcn_wmma_f32_16x16x4_f32(false, a, false, bn, (short)0, cn, false, false);
    }

    // extract diagonal: this lane holds column N=m; row M=r+8*hi, so diag when r=m-8*hi
    int r = m - 8 * hi;
    float loss = 0.0f;
    if (r >= 0 && r < 8) {
        float sp = cp[r] * 0.0625f;   // fold (1/(L+1))^2 = 1/16 into scores
        float sn = cn[r] * 0.0625f;
        float x = sn - sp;
        // stable softplus(x) = max(x,0) + log1p(exp(-|x|))
        loss = fmaxf(x, 0.0f) + log1pf(expf(-fabsf(x)));
    }
#pragma unroll
    for (int off = 16; off > 0; off >>= 1)
        loss += __shfl_down(loss, off, 32);
    if (lane == 0) partials[tile] = loss;
}

// Deterministic 512 -> 1 tree reduction; mean over BATCH
__global__ void lgcn_reduce(const float* __restrict__ partials, float* __restrict__ out) {
    __shared__ float s[512];
    int t = threadIdx.x;
    s[t] = partials[t];
    __syncthreads();
    for (int off = 256; off > 0; off >>= 1) {
        if (t < off) s[t] += s[t + off];
        __syncthreads();
    }
    if (t == 0) out[0] = s[0] * (1.0f / (float)BATCH);
}

extern "C" void kernel_launch(void* const* d_in, const int* in_sizes, int n_in,
                              void* d_out, int out_size, void* d_ws, size_t ws_size,
                              hipStream_t stream) {
    (void)n_in; (void)out_size; (void)ws_size;
    const float* ue    = (const float*)d_in[0];
    const float* ie    = (const float*)d_in[1];
    const int*   esrc  = (const int*)d_in[2];
    const int*   edst  = (const int*)d_in[3];
    const float* evals = (const float*)d_in[4];
    const int*   users = (const int*)d_in[5];
    const int*   pos   = (const int*)d_in[6];
    const int*   neg   = (const int*)d_in[7];
    const int E = in_sizes[2];   // 2 * N_INTERACTIONS directed edges

    float* acc      = (float*)d_ws;
    float* bufA     = acc + N64;
    float* bufB     = bufA + N64;
    float* partials = bufB + N64;   // 512 floats

    const int T = 256;
    const unsigned nVec = N64 / 4;
    lgcn_init<<<(nVec + T - 1) / T, T, 0, stream>>>(ue, ie, acc, bufA);

    float* cur = bufA;
    float* nxt = bufB;
    for (int l = 0; l < NLAYERS; ++l) {
        hipMemsetAsync(nxt, 0, (size_t)N64 * sizeof(float), stream);
        unsigned nThr = (unsigned)E * 16u;
        lgcn_scatter<<<(nThr + T - 1) / T, T, 0, stream>>>(esrc, edst, evals, cur, nxt, E);
        lgcn_accum<<<(nVec + T - 1) / T, T, 0, stream>>>(acc, nxt);
        float* tmp = cur; cur = nxt; nxt = tmp;
    }

    lgcn_score<<<BATCH / 16, 32, 0, stream>>>(acc, users, pos, neg, partials);
    lgcn_reduce<<<1, 512, 0, stream>>>(partials, (float*)d_out);
}